// DevNet_44839458570510
// MI455X (gfx1250) — compile-verified
//
#include <hip/hip_runtime.h>

typedef float v2f __attribute__((ext_vector_type(2)));
typedef float v4f __attribute__((ext_vector_type(4)));
typedef float v8f __attribute__((ext_vector_type(8)));
typedef unsigned int u32x4 __attribute__((ext_vector_type(4)));
typedef unsigned int u32x8 __attribute__((ext_vector_type(8)));

#define C_DIM 1152
#define CK 64                        // channels per K-tile
#define KTILES (C_DIM / CK)          // 18
#define ROWPAD 4                     // 16B pad per row (TDM pad_amount) -> conflict-free frags
#define LDS_STRIDE (CK + ROWPAD)     // 68 floats
#define WAVES_PER_BLOCK 4
#define TILE_M_PER_WAVE 16
#define TILE_M (WAVES_PER_BLOCK * TILE_M_PER_WAVE)  // 64 rows per block
#define N_PATCH 4096
#define B_BATCH 32
#define TOPK 409
#define T2 256                       // kernel-2 block size (8 waves)
#define EPT (N_PATCH / T2)           // 16 elements per thread

// ---------------------------------------------------------------------------
// One TDM descriptor per wave per K-tile: 2D tile, tile_dim0=64 elems (256B),
// tile_dim1=16 rows, tensor row stride 1152 elems.  LDS padding: 4 DWORDs
// after every 64 DWORDs (one row) -> LDS row stride 68 floats.
// D# packing per CDNA5 ISA.
// ---------------------------------------------------------------------------
__device__ __forceinline__ void tdm_load_tile(unsigned long long gaddr, unsigned lds_off)
{
    u32x4 g0;
    g0.x = 1u;                                    // count=1, is_restore=0, no gather
    g0.y = lds_off;                               // lds_addr (bytes)
    g0.z = (unsigned)gaddr;                       // global_addr[31:0]
    g0.w = ((unsigned)(gaddr >> 32) & 0x1FFFFFFu) // global_addr[56:32]
         | (2u << 30);                            // type = 2 (image)
    u32x8 g1;
    g1.s0 = (2u << 16)                            // data_size = 4 bytes
          | (1u << 20)                            // pad_enable
          | (5u << 22)                            // pad_interval: 2^5 * 8B = 256B = 64 DW
          | (3u << 25);                           // pad_amount: 4 DWORDs = 16B
    g1.s1 = ((unsigned)CK) << 16;                 // tensor_dim0[15:0]  (bits 63:48)
    g1.s2 = ((unsigned)TILE_M_PER_WAVE) << 16;    // tensor_dim1[15:0]  (bits 95:80)
    g1.s3 = ((unsigned)CK) << 16;                 // tile_dim0          (bits 127:112)
    g1.s4 = (unsigned)TILE_M_PER_WAVE;            // tile_dim1 (tile_dim2 = 0)
    g1.s5 = (unsigned)C_DIM;                      // tensor_dim0_stride[31:0] = 1152
    g1.s6 = 0u;                                   // stride0[47:32] | stride1[15:0]
    g1.s7 = 0u;                                   // stride1[47:16]
    asm volatile("tensor_load_to_lds %0, %1" :: "s"(g0), "s"(g1) : "memory");
}

// ---------------------------------------------------------------------------
// Kernel 1: scores[row] = dot(feature[row,:], W) + bias via WMMA f32 16x16x4.
//   A (16x4) = W chunk broadcast to all M rows (documented f32 A layout)
//   B (4x16) = 16 feature rows, column n = patch n -> D[:,n] = score of patch n
// Waves are independent after W preload: private double-buffered LDS tile,
// filled by TENSOR_LOAD_TO_LDS (TENSORcnt), consumed via ds_load_2addr_b64.
// ---------------------------------------------------------------------------
__global__ __launch_bounds__(128) void gemv_scores_wmma(
    const float* __restrict__ feature, const float* __restrict__ W,
    const float* __restrict__ bias, float* __restrict__ scores)
{
    __shared__ float ldsW[C_DIM];
    __shared__ float tile[WAVES_PER_BLOCK][2][TILE_M_PER_WAVE][LDS_STRIDE];

    const int t    = threadIdx.x;
    const int lane = t & 31;
    const int half = lane >> 4;        // 0: lanes 0-15 (K 0,1); 1: lanes 16-31 (K 2,3)
    const int mrow = lane & 15;

    // Preload W (1152 floats) into LDS once per block.
    for (int i = t; i < C_DIM; i += 128) ldsW[i] = W[i];
    __syncthreads();

    const int waveU    = __builtin_amdgcn_readfirstlane(t >> 5);
    const int wRowBase = blockIdx.x * TILE_M + waveU * TILE_M_PER_WAVE;

    const unsigned long long gbase =
        (unsigned long long)(uintptr_t)feature +
        (unsigned long long)wRowBase * (unsigned long long)(C_DIM * 4);
    // Low 32 bits of the generic address of a __shared__ object = LDS byte offset.
    unsigned ldsbuf0 = __builtin_amdgcn_readfirstlane(
        (unsigned)(uintptr_t)&tile[waveU][0][0][0]);
    unsigned ldsbuf1 = __builtin_amdgcn_readfirstlane(
        (unsigned)(uintptr_t)&tile[waveU][1][0][0]);

    const float bval = bias[0];

    v8f acc0 = {0.f,0.f,0.f,0.f,0.f,0.f,0.f,0.f};
    v8f acc1 = {0.f,0.f,0.f,0.f,0.f,0.f,0.f,0.f};

    tdm_load_tile(gbase, ldsbuf0);                 // prime buffer 0

    for (int kt = 0; kt < KTILES; ++kt) {
        const int buf = kt & 1;
        if (kt + 1 < KTILES) {
            // all fragment ds_loads from the buffer we are about to overwrite
            // must have landed in VGPRs before the TDM rewrites it
            asm volatile("s_wait_dscnt 0x0" ::: "memory");
            tdm_load_tile(gbase + (unsigned long long)(kt + 1) * (CK * 4),
                          buf ? ldsbuf0 : ldsbuf1);
            __builtin_amdgcn_s_wait_tensorcnt(1);  // tile kt landed, kt+1 in flight
        } else {
            __builtin_amdgcn_s_wait_tensorcnt(0);
        }
        asm volatile("" ::: "memory");             // compiler fence: loads stay below wait

        const int k0 = kt * CK;
        const float* frag = &tile[waveU][buf][mrow][half * 2];
        #pragma unroll
        for (int c = 0; c < CK; c += 8) {
            v2f a0 = *(const v2f*)&ldsW[k0 + c     + half * 2];
            v2f b0 = *(const v2f*)(frag + c);
            v2f a1 = *(const v2f*)&ldsW[k0 + c + 4 + half * 2];
            v2f b1 = *(const v2f*)(frag + c + 4);
            acc0 = __builtin_amdgcn_wmma_f32_16x16x4_f32(
                       false, a0, false, b0, (short)0, acc0, false, false);
            acc1 = __builtin_amdgcn_wmma_f32_16x16x4_f32(
                       false, a1, false, b1, (short)0, acc1, false, false);
        }
    }

    // D[m,n] = score[patch n] for every m; acc[0] on lanes 0..15 = (M=0, N=lane).
    const float s = acc0[0] + acc1[0] + bval;
    if (lane < 16)
        scores[wRowBase + lane] = s;
}

// ---------------------------------------------------------------------------
// Wave32 reductions: 5-step shfl_xor tree, one atomic per wave.
// ---------------------------------------------------------------------------
__device__ __forceinline__ int wave_sum_i32(int v) {
    #pragma unroll
    for (int m = 16; m >= 1; m >>= 1) v += __shfl_xor(v, m, 32);
    return v;
}
__device__ __forceinline__ float wave_sum_f32(float v) {
    #pragma unroll
    for (int m = 16; m >= 1; m >>= 1) v += __shfl_xor(v, m, 32);
    return v;
}

// ---------------------------------------------------------------------------
// Kernel 2: per batch (one 256-thread block), mean of top-409 |scores|.
// Binary search the 409th-largest value over the monotone uint encoding of
// non-negative floats, then  mean = (sum(v > kth) + (K - cnt_gt) * kth) / K.
// Counts are wave-reduced (shfl_xor) so only 8 LDS atomics per search step.
// ---------------------------------------------------------------------------
__global__ __launch_bounds__(T2) void topk_mean(
    const float* __restrict__ scores, float* __restrict__ out)
{
    __shared__ unsigned vals[N_PATCH];
    __shared__ int   cnt;
    __shared__ float fsum;
    __shared__ int   cgt;

    const int t    = threadIdx.x;
    const int lane = t & 31;
    const int b    = blockIdx.x;
    const float* s = scores + (size_t)b * N_PATCH;

    // Load |scores| bit patterns: coalesced b128 loads, 4 floats per pass.
    #pragma unroll
    for (int j = 0; j < EPT / 4; ++j) {           // 4 passes
        const int base = (j * T2 + t) * 4;
        v4f v = *(const v4f*)(s + base);
        u32x4 u;
        u.x = __float_as_uint(fabsf(v.x));
        u.y = __float_as_uint(fabsf(v.y));
        u.z = __float_as_uint(fabsf(v.z));
        u.w = __float_as_uint(fabsf(v.w));
        *(u32x4*)&vals[base] = u;
    }
    __syncthreads();

    unsigned lo = 0u, hi = 0x7F7FFFFFu;           // [0, max finite]
    while (lo < hi) {
        const unsigned mid = lo + ((hi - lo + 1u) >> 1);
        if (t == 0) cnt = 0;
        __syncthreads();
        int c = 0;
        #pragma unroll
        for (int j = 0; j < EPT; ++j)             // bank-conflict-free strided reads
            c += (vals[t + j * T2] >= mid) ? 1 : 0;
        c = wave_sum_i32(c);
        if (lane == 0) atomicAdd(&cnt, c);        // 8 atomics per step
        __syncthreads();
        const int total = cnt;
        __syncthreads();                          // all read cnt before next reset
        if (total >= TOPK) lo = mid; else hi = mid - 1u;
    }
    const unsigned kbits = lo;                    // bits of 409th-largest |score|
    const float    kval  = __uint_as_float(kbits);

    if (t == 0) { fsum = 0.f; cgt = 0; }
    __syncthreads();
    float ls = 0.f; int lc = 0;
    #pragma unroll
    for (int j = 0; j < EPT; ++j) {
        const unsigned u = vals[t + j * T2];
        if (u > kbits) { ls += __uint_as_float(u); ++lc; }
    }
    ls = wave_sum_f32(ls);
    lc = wave_sum_i32(lc);
    if (lane == 0) { atomicAdd(&fsum, ls); atomicAdd(&cgt, lc); }
    __syncthreads();
    if (t == 0)
        out[b] = (fsum + (float)(TOPK - cgt) * kval) / (float)TOPK;
}

// ---------------------------------------------------------------------------
extern "C" void kernel_launch(void* const* d_in, const int* in_sizes, int n_in,
                              void* d_out, int out_size, void* d_ws, size_t ws_size,
                              hipStream_t stream) {
    const float* feature = (const float*)d_in[0];   // [B, N, C] fp32
    const float* W       = (const float*)d_in[1];   // [C] fp32
    const float* bias    = (const float*)d_in[2];   // [1] fp32
    float* out    = (float*)d_out;                  // [B, 1] fp32
    float* scores = (float*)d_ws;                   // B*N floats = 512 KB scratch

    const int totalRows = B_BATCH * N_PATCH;        // 131072
    gemv_scores_wmma<<<totalRows / TILE_M, 128, 0, stream>>>(feature, W, bias, scores);
    topk_mean<<<B_BATCH, T2, 0, stream>>>(scores, out);
}